// SocialGraphAttention_46248207843654
// MI455X (gfx1250) — compile-verified
//
#include <hip/hip_runtime.h>
#include <cstdint>
#include <cstddef>

constexpr int BB    = 4;
constexpr int NN    = 2048;
constexpr int FF    = 256;
constexpr int HH    = 256;
constexpr int HEADS = 4;
constexpr int HD    = 64;
#define LN_EPS 1e-5f

typedef float v2f __attribute__((ext_vector_type(2)));
typedef float v8f __attribute__((ext_vector_type(8)));

static __device__ __forceinline__ v8f wmma_f32(v2f a, v2f b, v8f c) {
  // D = A(16x4) * B(4x16) + C(16x16), full fp32 matrix op (CDNA5 WMMA)
  return __builtin_amdgcn_wmma_f32_16x16x4_f32(false, a, false, b, (short)0, c,
                                               false, false);
}

static __device__ __forceinline__ float wred_max(float v) {
#pragma unroll
  for (int o = 16; o > 0; o >>= 1) v = fmaxf(v, __shfl_xor(v, o, 32));
  return v;
}
static __device__ __forceinline__ float wred_sum(float v) {
#pragma unroll
  for (int o = 16; o > 0; o >>= 1) v += __shfl_xor(v, o, 32);
  return v;
}

// ---------------------------------------------------------------------------
// Kernel A: f = nodes @ W_nodes + b_nodes   (8192 x 256 x 256, fp32 WMMA)
// One workgroup = 16 rows x 256 cols. 8 waves, 2 col-tiles each, K loop of 4.
// ---------------------------------------------------------------------------
__global__ __launch_bounds__(256) void k_node_linear(
    const float* __restrict__ nodes, const float* __restrict__ Wn,
    const float* __restrict__ bn, float* __restrict__ f_ws) {
  __shared__ float Atile[16][260];  // stride 260: conflict-free col reads, 16B rows
  const int r0  = blockIdx.x * 16;
  const int tid = threadIdx.x;

#pragma unroll
  for (int it = 0; it < 4; ++it) {          // 1024 float4 total
    int id  = tid + it * 256;
    int row = id >> 6, c4 = id & 63;
    float4 val = ((const float4*)(nodes + (size_t)(r0 + row) * FF))[c4];
    *(float4*)&Atile[row][c4 * 4] = val;
  }
  __syncthreads();

  const int w = tid >> 5, lane = tid & 31;
  const int halfsel = (lane < 16) ? 0 : 2;
  const int lrow    = lane & 15;
  const int n0      = w * 32;
  v8f c0 = {0, 0, 0, 0, 0, 0, 0, 0};
  v8f c1 = {0, 0, 0, 0, 0, 0, 0, 0};

  for (int k0 = 0; k0 < FF; k0 += 4) {
    v2f a;
    a.x = Atile[lrow][k0 + halfsel];
    a.y = Atile[lrow][k0 + halfsel + 1];
    const int kr = k0 + halfsel;
    v2f b0, b1;
    b0.x = Wn[(size_t)kr * HH + n0 + lrow];
    b0.y = Wn[(size_t)(kr + 1) * HH + n0 + lrow];
    b1.x = Wn[(size_t)kr * HH + n0 + 16 + lrow];
    b1.y = Wn[(size_t)(kr + 1) * HH + n0 + 16 + lrow];
    c0 = wmma_f32(a, b0, c0);
    c1 = wmma_f32(a, b1, c1);
  }

  const int mbase = (lane < 16) ? 0 : 8;
#pragma unroll
  for (int vv = 0; vv < 8; ++vv) {
    int m    = mbase + vv;
    int col0 = n0 + lrow, col1 = n0 + 16 + lrow;
    f_ws[(size_t)(r0 + m) * HH + col0] = c0[vv] + bn[col0];
    f_ws[(size_t)(r0 + m) * HH + col1] = c1[vv] + bn[col1];
  }
}

// ---------------------------------------------------------------------------
// Kernel B: s_i[b,h,n] = f[b,n,h,:].a1[h]; s_j likewise with a2.
// One wave per (b,n); 8 lanes per head; shuffle-reduce within 8-lane groups.
// ---------------------------------------------------------------------------
__global__ __launch_bounds__(256) void k_scores(
    const float* __restrict__ f_ws, const float* __restrict__ att_w,
    float* __restrict__ s_i, float* __restrict__ s_j) {
  const int tid = threadIdx.x;
  const int w = tid >> 5, lane = tid & 31;
  const int r = blockIdx.x * 8 + w;  // row in [0, B*N)
  const int b = r / NN, n = r % NN;
  const int h     = lane >> 3;
  const int dbase = (lane & 7) * 8;

  const float* frow = f_ws + (size_t)r * HH + h * HD;
  const float* a1   = att_w + (size_t)h * (2 * HD);
  const float* a2   = a1 + HD;
  float p1 = 0.f, p2 = 0.f;
#pragma unroll
  for (int k = 0; k < 8; ++k) {
    float fv = frow[dbase + k];
    p1 += fv * a1[dbase + k];
    p2 += fv * a2[dbase + k];
  }
#pragma unroll
  for (int o = 4; o > 0; o >>= 1) {
    p1 += __shfl_xor(p1, o, 32);
    p2 += __shfl_xor(p2, o, 32);
  }
  if ((lane & 7) == 0) {
    size_t idx = (size_t)(b * HEADS + h) * NN + n;
    s_i[idx] = p1;
    s_j[idx] = p2;
  }
}

// ---------------------------------------------------------------------------
// Kernel C: masked softmax + attn output + agg = attn_masked @ f_head (WMMA)
// One workgroup per (b, h, 16-row i-block). 8 waves.
// ---------------------------------------------------------------------------
__global__ __launch_bounds__(256) void k_attention(
    const float* __restrict__ f_ws, const int* __restrict__ adj,
    const float* __restrict__ s_i, const float* __restrict__ s_j,
    const float* __restrict__ att_b, float* __restrict__ attn_out,
    float* __restrict__ agg_ws) {
  __shared__ float sj_lds[NN];            // 8 KB
  __shared__ float si_lds[16];
  __shared__ float m_lds[16];
  __shared__ float den_lds[16];
  __shared__ float attn_tile[16][132];    // stride 132: conflict-free col reads
  __shared__ float f_tile[128][72];       // stride 72: half-wave banks disjoint
  __shared__ float agg_lds[16][64];

  const int bid  = blockIdx.x;
  const int iblk = bid & 127;
  const int bh   = bid >> 7;              // b*HEADS + h
  const int h    = bh & (HEADS - 1);
  const int b    = bh >> 2;
  const int i0   = iblk * 16;
  const int tid  = threadIdx.x;
  const int w = tid >> 5, lane = tid & 31;

  const float* sjrow = s_j + (size_t)bh * NN;
  const float* sirow = s_i + (size_t)bh * NN;
  const float  ab    = att_b[h];

  for (int k = 0; k < NN; k += 256) sj_lds[k + tid] = sjrow[k + tid];
  if (tid < 16) si_lds[tid] = sirow[i0 + tid];
#pragma unroll
  for (int k = 0; k < 4; ++k) ((float*)agg_lds)[tid + 256 * k] = 0.f;
  __syncthreads();

  // Pass 1: per-row softmax stats. Masked entries contribute score 0.0.
  for (int lr = 2 * w; lr < 2 * w + 2; ++lr) {
    const int   ig   = i0 + lr;
    const int*  arow = adj + (size_t)ig * NN;
    const float si   = si_lds[lr];
    float m = -3.4e38f;
    for (int j = lane; j < NN; j += 32) {
      float s = (arow[j] > 0) ? (si + sj_lds[j] + ab) : 0.f;
      m = fmaxf(m, s);
    }
    m = wred_max(m);
    float sum = 0.f;
    for (int j = lane; j < NN; j += 32) {
      float s = (arow[j] > 0) ? (si + sj_lds[j] + ab) : 0.f;
      sum += __expf(s - m);
    }
    sum = wred_sum(sum);
    if (lane == 0) { m_lds[lr] = m; den_lds[lr] = sum; }
  }
  __syncthreads();

  v8f acc[4];
#pragma unroll
  for (int t = 0; t < 4; ++t) acc[t] = (v8f){0, 0, 0, 0, 0, 0, 0, 0};
  const int halfsel = (lane < 16) ? 0 : 2;
  const int lrow16  = lane & 15;
  float* attn_base = attn_out + ((size_t)bh * NN + i0) * NN;

  for (int j0 = 0; j0 < NN; j0 += 128) {
    // stage f[b, j0:j0+128, h*64 : h*64+64] into LDS (coalesced float4)
#pragma unroll
    for (int it = 0; it < 8; ++it) {
      int id  = tid + it * 256;  // 2048 float4
      int row = id >> 4, c4 = id & 15;
      const float* src =
          f_ws + (size_t)(b * NN + j0 + row) * HH + h * HD + c4 * 4;
      float4 val = *(const float4*)src;
      *(float4*)&f_tile[row][c4 * 4] = val;
      if (j0 + 128 < NN) __builtin_prefetch(src + 128 * HH, 0, 0);
    }
    // attention values for this 16x128 tile (coalesced global attn writes)
    for (int lr = 2 * w; lr < 2 * w + 2; ++lr) {
      const int   ig   = i0 + lr;
      const int*  arow = adj + (size_t)ig * NN;
      const float si   = si_lds[lr];
      const float m    = m_lds[lr];
      const float rden = 1.f / den_lds[lr];
      for (int jj = lane; jj < 128; jj += 32) {
        int   j  = j0 + jj;
        int   a  = arow[j];
        float s  = (a > 0) ? (si + sj_lds[j] + ab) : 0.f;
        float av = __expf(s - m) * rden;
        attn_base[(size_t)lr * NN + j] = av;          // full softmax output
        attn_tile[lr][jj] = (a > 0) ? av : 0.f;       // re-masked for agg
      }
    }
    __syncthreads();

    // WMMA: wave w consumes local K slice [16w, 16w+16), 4 d-tiles of 16
    const int js = 16 * w;
#pragma unroll
    for (int kk = 0; kk < 4; ++kk) {
      const int jl = js + 4 * kk + halfsel;
      v2f a;
      a.x = attn_tile[lrow16][jl];
      a.y = attn_tile[lrow16][jl + 1];
#pragma unroll
      for (int t = 0; t < 4; ++t) {
        v2f bf;
        bf.x = f_tile[jl][16 * t + lrow16];
        bf.y = f_tile[jl + 1][16 * t + lrow16];
        acc[t] = wmma_f32(a, bf, acc[t]);
      }
    }
    __syncthreads();
  }

  // cross-wave reduction of partial 16x64 accumulators
  const int mbase = (lane < 16) ? 0 : 8;
#pragma unroll
  for (int t = 0; t < 4; ++t)
#pragma unroll
    for (int vv = 0; vv < 8; ++vv)
      atomicAdd(&agg_lds[mbase + vv][16 * t + lrow16], acc[t][vv]);
  __syncthreads();
#pragma unroll
  for (int k = 0; k < 4; ++k) {
    int idx = tid + 256 * k;
    int r = idx >> 6, c = idx & 63;
    agg_ws[(size_t)(b * NN + i0 + r) * HH + h * HD + c] = agg_lds[r][c];
  }
}

// ---------------------------------------------------------------------------
// Kernel D: y = agg @ W_out + b_out, then fused LayerNorm + ReLU.
// ---------------------------------------------------------------------------
__global__ __launch_bounds__(256) void k_out_ln(
    const float* __restrict__ agg_ws, const float* __restrict__ Wo,
    const float* __restrict__ bo, const float* __restrict__ gamma,
    const float* __restrict__ beta, float* __restrict__ out) {
  __shared__ float Atile[16][260];
  __shared__ float Ytile[16][260];
  const int r0  = blockIdx.x * 16;
  const int tid = threadIdx.x;

#pragma unroll
  for (int it = 0; it < 4; ++it) {
    int id  = tid + it * 256;
    int row = id >> 6, c4 = id & 63;
    float4 val = ((const float4*)(agg_ws + (size_t)(r0 + row) * HH))[c4];
    *(float4*)&Atile[row][c4 * 4] = val;
  }
  __syncthreads();

  const int w = tid >> 5, lane = tid & 31;
  const int halfsel = (lane < 16) ? 0 : 2;
  const int lrow    = lane & 15;
  const int n0      = w * 32;
  v8f c0 = {0, 0, 0, 0, 0, 0, 0, 0};
  v8f c1 = {0, 0, 0, 0, 0, 0, 0, 0};

  for (int k0 = 0; k0 < HH; k0 += 4) {
    v2f a;
    a.x = Atile[lrow][k0 + halfsel];
    a.y = Atile[lrow][k0 + halfsel + 1];
    const int kr = k0 + halfsel;
    v2f b0, b1;
    b0.x = Wo[(size_t)kr * HH + n0 + lrow];
    b0.y = Wo[(size_t)(kr + 1) * HH + n0 + lrow];
    b1.x = Wo[(size_t)kr * HH + n0 + 16 + lrow];
    b1.y = Wo[(size_t)(kr + 1) * HH + n0 + 16 + lrow];
    c0 = wmma_f32(a, b0, c0);
    c1 = wmma_f32(a, b1, c1);
  }

  const int mbase = (lane < 16) ? 0 : 8;
#pragma unroll
  for (int vv = 0; vv < 8; ++vv) {
    int m = mbase + vv;
    Ytile[m][n0 + lrow]      = c0[vv] + bo[n0 + lrow];
    Ytile[m][n0 + 16 + lrow] = c1[vv] + bo[n0 + 16 + lrow];
  }
  __syncthreads();

  for (int lr = 2 * w; lr < 2 * w + 2; ++lr) {
    float sum = 0.f, sq = 0.f;
#pragma unroll
    for (int k = 0; k < 8; ++k) {
      float y = Ytile[lr][lane + 32 * k];
      sum += y;
      sq += y * y;
    }
    sum = wred_sum(sum);
    sq  = wred_sum(sq);
    float mu   = sum * (1.f / HH);
    float var  = sq * (1.f / HH) - mu * mu;
    float rstd = rsqrtf(var + LN_EPS);
#pragma unroll
    for (int k = 0; k < 8; ++k) {
      int   c = lane + 32 * k;
      float y = (Ytile[lr][c] - mu) * rstd * gamma[c] + beta[c];
      out[(size_t)(r0 + lr) * HH + c] = fmaxf(y, 0.f);
    }
  }
}

// ---------------------------------------------------------------------------
extern "C" void kernel_launch(void* const* d_in, const int* in_sizes, int n_in,
                              void* d_out, int out_size, void* d_ws,
                              size_t ws_size, hipStream_t stream) {
  const float* nodes = (const float*)d_in[0];
  // d_in[1] = edges: unused by the reference
  const int*   adj   = (const int*)d_in[2];
  const float* Wn    = (const float*)d_in[3];
  const float* bn    = (const float*)d_in[4];
  const float* att_w = (const float*)d_in[5];
  const float* att_b = (const float*)d_in[6];
  const float* Wo    = (const float*)d_in[7];
  const float* bo    = (const float*)d_in[8];
  const float* gamma = (const float*)d_in[9];
  const float* beta  = (const float*)d_in[10];

  float* out  = (float*)d_out;
  float* attn = out + (size_t)BB * NN * HH;  // outputs concatenated flat

  float* f_ws = (float*)d_ws;                        // B*N*H
  float* s_i  = f_ws + (size_t)BB * NN * HH;         // B*HEADS*N
  float* s_j  = s_i + (size_t)BB * HEADS * NN;       // B*HEADS*N
  float* agg  = s_j + (size_t)BB * HEADS * NN;       // B*N*H

  k_node_linear<<<BB * NN / 16, 256, 0, stream>>>(nodes, Wn, bn, f_ws);
  k_scores<<<BB * NN / 8, 256, 0, stream>>>(f_ws, att_w, s_i, s_j);
  k_attention<<<BB * HEADS * (NN / 16), 256, 0, stream>>>(f_ws, adj, s_i, s_j,
                                                          att_b, attn, agg);
  k_out_ln<<<BB * NN / 16, 256, 0, stream>>>(agg, Wo, bo, gamma, beta, out);
}